// GAT_50422916055441
// MI455X (gfx1250) — compile-verified
//
#include <hip/hip_runtime.h>
#include <climits>

#define FDIM 128          // feature dim (both in and out of every GEMM here)
#define HEADS 2
#define CH 64
#define NEG_SLOPE 0.2f

typedef float v2f __attribute__((ext_vector_type(2)));
typedef float v8f __attribute__((ext_vector_type(8)));

// ---------------------------------------------------------------------------
// GEMM: OUT[N,128] = X[N,128] @ W[128,128] (+ bias0 + bias1 broadcast per col)
// One wave per 16-row strip; 8 col-tiles of 16; K in steps of 4 (f32 WMMA).
// ---------------------------------------------------------------------------
__global__ __launch_bounds__(128) void gemm_f32_wmma(
    const float* __restrict__ X, const float* __restrict__ W,
    const float* __restrict__ bias0, const float* __restrict__ bias1,
    float* __restrict__ OUT, int nrows)
{
    const int ntiles = nrows >> 4;
    const int tile = blockIdx.x * 4 + (threadIdx.x >> 5);
    if (tile >= ntiles) return;

    const int lane  = threadIdx.x & 31;
    const int lhalf = lane >> 4;   // 0: K sub {0,1}, 1: K sub {2,3}
    const int lidx  = lane & 15;   // M index for A; N index for B/C/D
    const int row0  = tile * 16;

    // A fragment source: lane holds X[row0+lidx][kk + 2*lhalf + {0,1}]
    const float* xrow = X + (size_t)(row0 + lidx) * FDIM + 2 * lhalf;

    v8f acc[8];
#pragma unroll
    for (int t = 0; t < 8; ++t)
#pragma unroll
        for (int r = 0; r < 8; ++r) acc[t][r] = 0.0f;

#pragma unroll 4
    for (int kk = 0; kk < FDIM; kk += 4) {
        v2f a = *(const v2f*)(xrow + kk);          // aligned b64 load
        const int krow = kk + 2 * lhalf;
        const float* wr0 = W + (size_t)krow * FDIM + lidx;
        const float* wr1 = wr0 + FDIM;
#pragma unroll
        for (int t = 0; t < 8; ++t) {
            v2f b;
            b.x = wr0[t * 16];                     // B[k=krow  ][n=t*16+lidx]
            b.y = wr1[t * 16];                     // B[k=krow+1][n=t*16+lidx]
            acc[t] = __builtin_amdgcn_wmma_f32_16x16x4_f32(
                false, a, false, b, (short)0, acc[t], false, false);
        }
    }

    // C/D layout: lanes 0-15 -> M=r, lanes 16-31 -> M=8+r; N = lidx
    const int orow = row0 + 8 * lhalf;
#pragma unroll
    for (int t = 0; t < 8; ++t) {
        const int col = t * 16 + lidx;
        float badd = 0.0f;
        if (bias0) badd += bias0[col];
        if (bias1) badd += bias1[col];
        float* o = OUT + (size_t)orow * FDIM + col;
#pragma unroll
        for (int r = 0; r < 8; ++r)
            o[(size_t)r * FDIM] = acc[t][r] + badd;
    }
}

// ---------------------------------------------------------------------------
// a_s[n,h] = <Hp[n,h,:], att_src[h,:]> ; a_d likewise. One thread per (n,h).
// ---------------------------------------------------------------------------
__global__ void att_kernel(const float* __restrict__ Hp,
                           const float* __restrict__ att_src,
                           const float* __restrict__ att_dst,
                           float* __restrict__ a_s, float* __restrict__ a_d,
                           int n)
{
    int i = blockIdx.x * blockDim.x + threadIdx.x;
    if (i >= n * HEADS) return;
    const int node = i >> 1, h = i & 1;
    const float* hp = Hp + (size_t)node * FDIM + h * CH;
    const float* as = att_src + h * CH;
    const float* ad = att_dst + h * CH;
    float s0 = 0.f, s1 = 0.f;
#pragma unroll 8
    for (int c = 0; c < CH; ++c) {
        float v = hp[c];
        s0 += v * as[c];
        s1 += v * ad[c];
    }
    a_s[i] = s0;
    a_d[i] = s1;
}

__global__ void init_kernel(int* __restrict__ m, float* __restrict__ denom, int n2)
{
    int i = blockIdx.x * blockDim.x + threadIdx.x;
    if (i < n2) { m[i] = INT_MIN; denom[i] = 0.0f; }
}

__device__ __forceinline__ int   fmapf(float x)  { int b = __float_as_int(x); return b >= 0 ? b : (b ^ 0x7FFFFFFF); }
__device__ __forceinline__ float funmap(int b)   { return __int_as_float(b >= 0 ? b : (b ^ 0x7FFFFFFF)); }
__device__ __forceinline__ float lrelu(float x)  { return x > 0.f ? x : NEG_SLOPE * x; }

// entries k < E are graph edges, k >= E are self loops (s = d = k - E)
__global__ void edge_max_kernel(const int* __restrict__ src, const int* __restrict__ dst,
                                const float* __restrict__ a_s, const float* __restrict__ a_d,
                                int* __restrict__ m, int E_, int M_)
{
    int k = blockIdx.x * blockDim.x + threadIdx.x;
    if (k >= M_) return;
    const int s = k < E_ ? src[k] : k - E_;
    const int d = k < E_ ? dst[k] : k - E_;
#pragma unroll
    for (int h = 0; h < HEADS; ++h) {
        float e = lrelu(a_s[s * HEADS + h] + a_d[d * HEADS + h]);
        atomicMax(&m[d * HEADS + h], fmapf(e));
    }
}

__global__ void edge_sum_kernel(const int* __restrict__ src, const int* __restrict__ dst,
                                const float* __restrict__ a_s, const float* __restrict__ a_d,
                                const int* __restrict__ m, float* __restrict__ P,
                                float* __restrict__ denom, int E_, int M_)
{
    int k = blockIdx.x * blockDim.x + threadIdx.x;
    if (k >= M_) return;
    const int s = k < E_ ? src[k] : k - E_;
    const int d = k < E_ ? dst[k] : k - E_;
#pragma unroll
    for (int h = 0; h < HEADS; ++h) {
        float e  = lrelu(a_s[s * HEADS + h] + a_d[d * HEADS + h]);
        float mx = funmap(m[d * HEADS + h]);
        float p  = __expf(e - mx);
        P[(size_t)k * HEADS + h] = p;
        atomicAdd(&denom[d * HEADS + h], p);
    }
}

// one wave per entry; lane handles 4 consecutive channels (float4 + 4 atomics)
__global__ __launch_bounds__(256) void edge_scatter_kernel(
    const int* __restrict__ src, const int* __restrict__ dst,
    const float* __restrict__ P, const float* __restrict__ denom,
    const float* __restrict__ Hp, float* __restrict__ out, int E_, int M_)
{
    const int lane = threadIdx.x & 31;
    const int k = blockIdx.x * 8 + (threadIdx.x >> 5);
    if (k >= M_) return;
    const int s = k < E_ ? src[k] : k - E_;
    const int d = k < E_ ? dst[k] : k - E_;
    const int h = lane >> 4;                       // lanes 0-15 head 0, 16-31 head 1
    const float alpha = P[(size_t)k * HEADS + h] / (denom[d * HEADS + h] + 1e-16f);
    const int c0 = lane * 4;
    const float4 hv = *(const float4*)(Hp + (size_t)s * FDIM + c0);
    float* o = out + (size_t)d * FDIM + c0;
    atomicAdd(o + 0, hv.x * alpha);
    atomicAdd(o + 1, hv.y * alpha);
    atomicAdd(o + 2, hv.z * alpha);
    atomicAdd(o + 3, hv.w * alpha);
}

__global__ void relu_kernel(float* __restrict__ x, int n)
{
    int i = blockIdx.x * blockDim.x + threadIdx.x;
    if (i < n) x[i] = fmaxf(x[i], 0.0f);
}

// ---------------------------------------------------------------------------
extern "C" void kernel_launch(void* const* d_in, const int* in_sizes, int n_in,
                              void* d_out, int out_size, void* d_ws, size_t ws_size,
                              hipStream_t stream)
{
    const float* x        = (const float*)d_in[0];
    const int*   src      = (const int*)  d_in[1];
    const int*   dst      = (const int*)  d_in[2];
    const float* W1       = (const float*)d_in[3];
    const float* att_src1 = (const float*)d_in[4];
    const float* att_dst1 = (const float*)d_in[5];
    const float* b1       = (const float*)d_in[6];
    const float* Wsk1     = (const float*)d_in[7];
    const float* bsk1     = (const float*)d_in[8];
    const float* W2       = (const float*)d_in[9];
    const float* att_src2 = (const float*)d_in[10];
    const float* att_dst2 = (const float*)d_in[11];
    const float* b2       = (const float*)d_in[12];
    const float* Wsk2     = (const float*)d_in[13];
    const float* bsk2     = (const float*)d_in[14];

    const int N  = in_sizes[0] / FDIM;     // 50000
    const int E  = in_sizes[1];            // 800000
    const int M  = E + N;                  // entries incl. self loops
    float* out   = (float*)d_out;

    // workspace layout (256B-aligned chunks)
    char* base = (char*)d_ws;
    size_t off = 0;
    auto alloc = [&](size_t bytes) {
        void* p = base + off;
        off = (off + bytes + 255) & ~(size_t)255;
        return p;
    };
    float* Hproj = (float*)alloc((size_t)N * FDIM * 4);   // x@W projection
    float* Hbuf  = (float*)alloc((size_t)N * FDIM * 4);   // layer-1 output / layer-2 input
    float* a_s   = (float*)alloc((size_t)N * HEADS * 4);
    float* a_d   = (float*)alloc((size_t)N * HEADS * 4);
    int*   mbuf  = (int*)  alloc((size_t)N * HEADS * 4);
    float* denom = (float*)alloc((size_t)N * HEADS * 4);
    float* P     = (float*)alloc((size_t)M * HEADS * 4);

    const int ntiles     = N / 16;                         // 3125
    const dim3 gGemm((ntiles + 3) / 4), bGemm(128);
    const dim3 gAtt((N * HEADS + 255) / 256), b256(256);
    const dim3 gInit((N * HEADS + 255) / 256);
    const dim3 gEdge((M + 255) / 256);
    const dim3 gScat((M + 7) / 8);
    const dim3 gRelu((N * FDIM + 255) / 256);

    // ---------------- layer 1 ----------------
    gemm_f32_wmma<<<gGemm, bGemm, 0, stream>>>(x, W1, nullptr, nullptr, Hproj, N);
    gemm_f32_wmma<<<gGemm, bGemm, 0, stream>>>(x, Wsk1, b1, bsk1, Hbuf, N);   // skip path + both biases
    att_kernel<<<gAtt, b256, 0, stream>>>(Hproj, att_src1, att_dst1, a_s, a_d, N);
    init_kernel<<<gInit, b256, 0, stream>>>(mbuf, denom, N * HEADS);
    edge_max_kernel<<<gEdge, b256, 0, stream>>>(src, dst, a_s, a_d, mbuf, E, M);
    edge_sum_kernel<<<gEdge, b256, 0, stream>>>(src, dst, a_s, a_d, mbuf, P, denom, E, M);
    edge_scatter_kernel<<<gScat, b256, 0, stream>>>(src, dst, P, denom, Hproj, Hbuf, E, M);
    relu_kernel<<<gRelu, b256, 0, stream>>>(Hbuf, N * FDIM);

    // ---------------- layer 2 ----------------
    gemm_f32_wmma<<<gGemm, bGemm, 0, stream>>>(Hbuf, W2, nullptr, nullptr, Hproj, N);
    gemm_f32_wmma<<<gGemm, bGemm, 0, stream>>>(Hbuf, Wsk2, b2, bsk2, out, N);
    att_kernel<<<gAtt, b256, 0, stream>>>(Hproj, att_src2, att_dst2, a_s, a_d, N);
    init_kernel<<<gInit, b256, 0, stream>>>(mbuf, denom, N * HEADS);
    edge_max_kernel<<<gEdge, b256, 0, stream>>>(src, dst, a_s, a_d, mbuf, E, M);
    edge_sum_kernel<<<gEdge, b256, 0, stream>>>(src, dst, a_s, a_d, mbuf, P, denom, E, M);
    edge_scatter_kernel<<<gScat, b256, 0, stream>>>(src, dst, P, denom, Hproj, out, E, M);
}